// SimpleAttentionModel_18150531793223
// MI455X (gfx1250) — compile-verified
//
#include <hip/hip_runtime.h>
#include <math.h>

// Problem constants (from reference)
#define BB 8
#define SS 4096
#define DD 256
#define HH 1024
#define VV 32000
#define NCHUNK 64
#define SPC (SS / NCHUNK)   // 64 sequence positions per chunk

typedef __attribute__((ext_vector_type(2))) float v2f;
typedef __attribute__((ext_vector_type(8))) float v8f;

// ---------------------------------------------------------------------------
// K1: partial embedding sums  epart[b][c][d] = sum_{s in chunk c} emb[x[b,s]][d]
// (PAD token 0 rows are skipped == reference's zeroed padding row)
// ---------------------------------------------------------------------------
__global__ void k_esum_part(const int* __restrict__ x, const float* __restrict__ emb,
                            float* __restrict__ epart) {
  const int b = blockIdx.x, c = blockIdx.y, d = threadIdx.x;
  float acc = 0.f;
  const int sbase = c * SPC;
  for (int i = 0; i < SPC; ++i) {
    const int tok = x[b * SS + sbase + i];
    if (tok != 0) acc += emb[(size_t)tok * DD + d];
  }
  epart[((size_t)b * NCHUNK + c) * DD + d] = acc;
}

// ---------------------------------------------------------------------------
// K2: esum reduce -> qsum = esum @ wq^T -> r = qsum @ wk   (one block per b)
// ---------------------------------------------------------------------------
__global__ void k_qsum_r(const float* __restrict__ epart, const float* __restrict__ wq,
                         const float* __restrict__ wk, float* __restrict__ r) {
  __shared__ float es[DD];
  __shared__ float qs[DD];
  const int b = blockIdx.x, t = threadIdx.x;
  float acc = 0.f;
  for (int c = 0; c < NCHUNK; ++c) acc += epart[((size_t)b * NCHUNK + c) * DD + t];
  es[t] = acc;
  __syncthreads();
  // qsum[e=t] = sum_d es[d] * wq[t][d]
  float q = 0.f;
  const float* wr = wq + (size_t)t * DD;
  for (int d = 0; d < DD; ++d) q += es[d] * wr[d];
  qs[t] = q;
  __syncthreads();
  // r[d=t] = sum_e qs[e] * wk[e][t]   (coalesced over t)
  float rv = 0.f;
  for (int e = 0; e < DD; ++e) rv += qs[e] * wk[(size_t)e * DD + t];
  r[b * DD + t] = rv;
}

// ---------------------------------------------------------------------------
// K3: second embedding pass: a_s = e_s . r[b];  tpart += a_s * e_s
// One wave per position (token is wave-uniform), shfl_xor dot reduction.
// ---------------------------------------------------------------------------
__global__ void k_tpart(const int* __restrict__ x, const float* __restrict__ emb,
                        const float* __restrict__ r, float* __restrict__ tpart) {
  __shared__ float rs[DD];
  __shared__ float tred[8 * DD];
  const int b = blockIdx.x, c = blockIdx.y, t = threadIdx.x;
  rs[t] = r[b * DD + t];
  __syncthreads();
  const int wave = t >> 5, lane = t & 31;
  const int dbase = lane * 8;
  float acc[8];
#pragma unroll
  for (int j = 0; j < 8; ++j) acc[j] = 0.f;
  for (int i = 0; i < SPC / 8; ++i) {
    const int s = c * SPC + i * 8 + wave;
    const int tok = x[b * SS + s];          // uniform across the wave
    if (tok != 0) {
      const float* ep = emb + (size_t)tok * DD + dbase;
      float e[8];
#pragma unroll
      for (int j = 0; j < 8; ++j) e[j] = ep[j];
      float p = 0.f;
#pragma unroll
      for (int j = 0; j < 8; ++j) p += e[j] * rs[dbase + j];
#pragma unroll
      for (int m = 16; m > 0; m >>= 1) p += __shfl_xor(p, m, 32);
#pragma unroll
      for (int j = 0; j < 8; ++j) acc[j] += p * e[j];
    }
  }
#pragma unroll
  for (int j = 0; j < 8; ++j) tred[wave * DD + dbase + j] = acc[j];
  __syncthreads();
  float sum = 0.f;
  for (int w = 0; w < 8; ++w) sum += tred[w * DD + t];
  tpart[((size_t)b * NCHUNK + c) * DD + t] = sum;
}

// ---------------------------------------------------------------------------
// K4: t reduce -> v_out = t @ wv^T -> h = relu(v_out @ w1^T + b1)
// Also zeroes the M-padding rows 8..15 of h for the WMMA kernel.
// ---------------------------------------------------------------------------
__global__ void k_vout_h(const float* __restrict__ tpart, const float* __restrict__ wv,
                         const float* __restrict__ w1, const float* __restrict__ b1,
                         float* __restrict__ h) {
  __shared__ float ts[DD];
  __shared__ float vs[DD];
  const int b = blockIdx.x, t = threadIdx.x;
  float acc = 0.f;
  for (int c = 0; c < NCHUNK; ++c) acc += tpart[((size_t)b * NCHUNK + c) * DD + t];
  ts[t] = acc;
  __syncthreads();
  float vo = 0.f;
  const float* wr = wv + (size_t)t * DD;
  for (int d = 0; d < DD; ++d) vo += ts[d] * wr[d];
  vs[t] = vo;
  __syncthreads();
  for (int j = t; j < HH; j += 256) {
    float hv = b1[j];
    const float* w1r = w1 + (size_t)j * DD;
    for (int d = 0; d < DD; ++d) hv += vs[d] * w1r[d];
    h[(size_t)b * HH + j] = fmaxf(hv, 0.f);
    h[(size_t)(b + BB) * HH + j] = 0.f;   // zero M-padding rows
  }
}

// ---------------------------------------------------------------------------
// K5: logits[16(pad) x 32000] = h[16 x 1024] @ w2^T via V_WMMA_F32_16X16X4_F32.
// One wave per 16-wide vocab tile; block of 8 waves covers 128 rows of w2
// (streamed exactly once). fp32 WMMA => same precision as the fp32 reference.
// A layout (16x4 f32): lanes 0-15 hold K={0,1}, lanes 16-31 hold K={2,3};
// B layout (4x16 f32) mirrors it, so both operands use offset k + 2*half.
// ---------------------------------------------------------------------------
__global__ void k_logits_wmma(const float* __restrict__ h, const float* __restrict__ w2,
                              const float* __restrict__ b2, float* __restrict__ logits) {
  const int wave = threadIdx.x >> 5, lane = threadIdx.x & 31;
  const int n0 = blockIdx.x * 128 + wave * 16;
  const int half = lane >> 4, nl = lane & 15;
  const float* arow = h + (size_t)nl * HH + 2 * half;          // A: h[M=nl][k..]
  const float* brow = w2 + (size_t)(n0 + nl) * HH + 2 * half;  // B: w2[N=n0+nl][k..]
  v8f c = {};
#pragma unroll 8
  for (int k = 0; k < HH; k += 4) {
    if ((k & 127) == 0) __builtin_prefetch(brow + k + 128, 0, 0);
    v2f a = *(const v2f*)(arow + k);
    v2f bv = *(const v2f*)(brow + k);
    c = __builtin_amdgcn_wmma_f32_16x16x4_f32(false, a, false, bv,
                                              (short)0, c, false, false);
  }
  if (half == 0) {  // lanes 0-15 hold D rows M=0..7 (real batches) in c[0..7]
    const float bias = b2[n0 + nl];
#pragma unroll
    for (int j = 0; j < BB; ++j)
      logits[(size_t)j * VV + n0 + nl] = c[j] + bias;
  }
}

// ---------------------------------------------------------------------------
// K6: row softmax over 32000 (one block per batch row; logits hot in L2)
// ---------------------------------------------------------------------------
__global__ void k_softmax(const float* __restrict__ logits, float* __restrict__ out) {
  __shared__ float red[1024];
  const int b = blockIdx.x, t = threadIdx.x;
  const float* row = logits + (size_t)b * VV;
  float m = -INFINITY;
  for (int v = t; v < VV; v += 1024) m = fmaxf(m, row[v]);
  red[t] = m;
  __syncthreads();
  for (int off = 512; off > 0; off >>= 1) {
    if (t < off) red[t] = fmaxf(red[t], red[t + off]);
    __syncthreads();
  }
  const float mx = red[0];
  __syncthreads();
  float s = 0.f;
  for (int v = t; v < VV; v += 1024) s += expf(row[v] - mx);
  red[t] = s;
  __syncthreads();
  for (int off = 512; off > 0; off >>= 1) {
    if (t < off) red[t] += red[t + off];
    __syncthreads();
  }
  const float inv = 1.f / red[0];
  for (int v = t; v < VV; v += 1024) out[(size_t)b * VV + v] = expf(row[v] - mx) * inv;
}

// ---------------------------------------------------------------------------
extern "C" void kernel_launch(void* const* d_in, const int* in_sizes, int n_in,
                              void* d_out, int out_size, void* d_ws, size_t ws_size,
                              hipStream_t stream) {
  const int*   x   = (const int*)  d_in[0];
  const float* emb = (const float*)d_in[1];
  const float* wq  = (const float*)d_in[2];
  const float* wk  = (const float*)d_in[3];
  const float* wv  = (const float*)d_in[4];
  const float* w1  = (const float*)d_in[5];
  const float* b1  = (const float*)d_in[6];
  const float* w2  = (const float*)d_in[7];
  const float* b2  = (const float*)d_in[8];
  float* out = (float*)d_out;

  // Workspace layout (floats). 'part' is reused by K1/K2 then K3/K4.
  float* ws     = (float*)d_ws;
  float* part   = ws;                               // [8][64][256]   = 131072
  float* r      = part + (size_t)BB * NCHUNK * DD;  // [8][256]       = 2048
  float* h      = r + (size_t)BB * DD;              // [16][1024]     = 16384
  float* logits = h + (size_t)2 * BB * HH;          // [8][32000]     = 256000

  k_esum_part  <<<dim3(BB, NCHUNK), 256, 0, stream>>>(x, emb, part);
  k_qsum_r     <<<BB,               256, 0, stream>>>(part, wq, wk, r);
  k_tpart      <<<dim3(BB, NCHUNK), 256, 0, stream>>>(x, emb, r, part);
  k_vout_h     <<<BB,               256, 0, stream>>>(part, wv, w1, b1, h);
  k_logits_wmma<<<VV / 128,         256, 0, stream>>>(h, w2, b2, logits);
  k_softmax    <<<BB,              1024, 0, stream>>>(logits, out);
}